// Embedding_5557687681682
// MI455X (gfx1250) — compile-verified
//
#include <hip/hip_runtime.h>

// out[b, v, n] = x[b, v] * W[v, n] + bias[v, n]
// B=8192, V=64, N=128.  Store-bandwidth-bound: 256 MB output, ~11 us floor
// @ 23.3 TB/s HBM; the 134 MFLOP are free.  Compute routed through
// V_WMMA_F32_16X16X4_F32 using two of the four K slots:
//     K=0:  A = x[m],  B = W[n]
//     K=1:  A = 1.0,   B = bias[n]      =>  D = x*W + bias,  C = inline 0
// Output stored non-temporal (256 MB > 192 MB L2, never re-read).
//
// One wave32 per (16-batch-row, v) pair: A built once, 8 WMMAs across the
// 128-wide n dimension, 64 stores sharing one base + immediate offsets
// (row stride 32 KB < 2^23).

typedef __attribute__((ext_vector_type(2))) float v2f;
typedef __attribute__((ext_vector_type(8))) float v8f;

#define BATCH 8192
#define VEC   64
#define NN    128

__global__ __launch_bounds__(256) void fused_dense_stack_wmma(
    const float* __restrict__ x,     // [BATCH, VEC]
    const float* __restrict__ W,     // [VEC, NN]
    const float* __restrict__ bias,  // [VEC, NN]
    float* __restrict__ out)         // [BATCH, VEC, NN]
{
    const int wave = (int)((blockIdx.x * blockDim.x + threadIdx.x) >> 5);
    const int lane = (int)(threadIdx.x & 31);

    const int v  = wave & 63;         // vector dim: 0..63
    const int bt = wave >> 6;         // batch tile: 0..511 (BATCH/16)

    const int laneN = lane & 15;      // column / row-in-tile index
    const int hi    = lane >> 4;      // 0: lanes 0-15, 1: lanes 16-31
    const bool lo   = (hi == 0);

    // A (16x4 f32, 2 VGPRs): lanes 0-15 hold K=0 (VGPR0) and K=1 (VGPR1);
    // lanes 16-31 hold K=2, K=3 (kept zero).  Load at lane&15 for all lanes
    // (always in-bounds), mask by value -> v_cndmask, EXEC stays all-ones.
    const float xv = x[(bt * 16 + laneN) * VEC + v];
    v2f a;
    a.x = lo ? xv   : 0.0f;   // K=0 column: x[m]
    a.y = lo ? 1.0f : 0.0f;   // K=1 column: 1.0 (carries the bias row of B)

    // Single store base; every store offset below is a compile-time constant.
    float* const p = out + ((size_t)(bt * 16 + hi * 8) * VEC + v) * NN + laneN;

    const float* const wrow = W    + v * NN + laneN;
    const float* const brow = bias + v * NN + laneN;

#pragma unroll
    for (int nt = 0; nt < 8; ++nt) {
        // B (4x16 f32, 2 VGPRs): row K=0 -> VGPR0 lanes 0-15 (W),
        //                        row K=1 -> VGPR1 lanes 0-15 (bias).
        const float wv = wrow[nt * 16];
        const float bb = brow[nt * 16];
        v2f bm;
        bm.x = lo ? wv : 0.0f;
        bm.y = lo ? bb : 0.0f;

        // C = 0 (loop-invariant; SRC2 can be inline 0 per ISA).
        v8f c = {};

        // D = A x B + 0 = x*W + bias   (v_wmma_f32_16x16x4_f32)
        v8f d = __builtin_amdgcn_wmma_f32_16x16x4_f32(
            /*neg_a=*/false, a, /*neg_b=*/false, bm,
            /*c_mod=*/(short)0, c, /*reuse_a=*/false, /*reuse_b=*/false);

        // Store tile: VGPR r -> rows M=r (lanes 0-15) / M=r+8 (lanes 16-31).
        // Offsets: nt*16 floats within the row, r*VEC*NN floats between rows.
#pragma unroll
        for (int r = 0; r < 8; ++r) {
            __builtin_nontemporal_store(d[r], p + nt * 16 + r * (VEC * NN));
        }
    }
}

extern "C" void kernel_launch(void* const* d_in, const int* in_sizes, int n_in,
                              void* d_out, int out_size, void* d_ws, size_t ws_size,
                              hipStream_t stream) {
    (void)in_sizes; (void)n_in; (void)out_size; (void)d_ws; (void)ws_size;
    const float* x    = (const float*)d_in[0];  // [8192, 64]
    const float* W    = (const float*)d_in[1];  // [64, 128]
    const float* bias = (const float*)d_in[2];  // [64, 128]
    float* out        = (float*)d_out;          // [8192, 64, 128]

    const int totalWaves = (BATCH / 16) * VEC;      // 32768 waves
    const int threads = 256;                        // 8 waves/block
    const int blocks  = totalWaves * 32 / threads;  // 4096

    fused_dense_stack_wmma<<<blocks, threads, 0, stream>>>(x, W, bias, out);
}